// Caps_Layer_16870631538762
// MI455X (gfx1250) — compile-verified
//
#include <hip/hip_runtime.h>
#include <hip/hip_bf16.h>
#include <math.h>

// CapsNet layer for MI455X (gfx1250, wave32).
//   u_hat = X[16384,768] @ W[768,512]  (fp32, V_WMMA_F32_16X16X4_F32)
//   3 rounds of dynamic routing (softmax over 16 capsules, reductions, L2-norm)
//
// d_ws layout: u_hat as float[B*S*N*C] = float[16384*512] = 33.5 MB.

#define B_DIM 32
#define S_DIM 512
#define D_DIM 768
#define NCAP  16
#define CCAP  32
#define NC    (NCAP * CCAP)      // 512
#define M_TOT (B_DIM * S_DIM)    // 16384

typedef float v2f __attribute__((ext_vector_type(2)));
typedef float v8f __attribute__((ext_vector_type(8)));

// ---------------------------------------------------------------------------
// GEMM: u_hat[M_TOT, NC] = X[M_TOT, D] * W[D, NC]   (all fp32, row-major)
// Workgroup: 256 threads = 8 waves. Macro-tile 128(M) x 64(N), K chunk = 32,
// double-buffered LDS. Wave (wm, wn) computes a 32x32 tile = 4 WMMA tiles.
// W tile stored K-pair interleaved so every B fragment is one 8B LDS load.
// ---------------------------------------------------------------------------
#define KC 32
#define MT 128
#define NT 64
#define XS_STRIDE 34            // pad X tile rows (bank spread, 8B-aligned rows)
#define WP_ROW (NT * 2)         // 128 floats per k-pair row

__global__ __launch_bounds__(256)
void caps_gemm_wmma_f32(const float* __restrict__ X,
                        const float* __restrict__ W,
                        float* __restrict__ U) {
    __shared__ float Xs[2][MT * XS_STRIDE];      // 2 x 17408 B
    __shared__ float Wp[2][(KC / 2) * WP_ROW];   // 2 x  8192 B

    const int t    = threadIdx.x;
    const int wave = t >> 5;
    const int lane = t & 31;
    const int h    = lane >> 4;   // half-wave: K pair {0,1} vs {2,3}
    const int r    = lane & 15;   // row (A) / column (B) within 16
    const int wm   = wave & 3;    // 4 M-blocks of 32
    const int wn   = wave >> 2;   // 2 N-blocks of 32

    const int m0 = blockIdx.y * MT;   // 128 blocks
    const int n0 = blockIdx.x * NT;   //   8 blocks

    v8f a00 = {0.f,0.f,0.f,0.f,0.f,0.f,0.f,0.f};
    v8f a01 = a00, a10 = a00, a11 = a00;

    // Stage K-chunk k0c into LDS buffer bb (global -> regs -> LDS).
    auto stage = [&](int k0c, int bb) {
        // X tile: 128 rows x 32 K; four float4 loads per thread.
        #pragma unroll
        for (int i = 0; i < 4; ++i) {
            int idx  = t + i * 256;          // 0..1023, 8 float4 per row
            int row  = idx >> 3;
            int col4 = idx & 7;
            const float4 g =
                *(const float4*)&X[(size_t)(m0 + row) * D_DIM + k0c + col4 * 4];
            float* dst = &Xs[bb][row * XS_STRIDE + col4 * 4];
            *(float2*)(dst)     = make_float2(g.x, g.y);   // 8B-aligned (stride 34)
            *(float2*)(dst + 2) = make_float2(g.z, g.w);
        }
        // W tile: 32 K x 64 N, stored K-pair interleaved:
        //   Wp[kp][n] = { W[2kp][n], W[2kp+1][n] }   (float2 per element)
        {
            int kp = t >> 4;                 // 0..15
            int n4 = t & 15;                 // 0..15 -> 4 columns each
            const float4 r0 =
                *(const float4*)&W[(size_t)(k0c + 2 * kp)     * NC + n0 + n4 * 4];
            const float4 r1 =
                *(const float4*)&W[(size_t)(k0c + 2 * kp + 1) * NC + n0 + n4 * 4];
            float4 lo = make_float4(r0.x, r1.x, r0.y, r1.y);
            float4 hi = make_float4(r0.z, r1.z, r0.w, r1.w);
            *(float4*)&Wp[bb][kp * WP_ROW + n4 * 8]     = lo;
            *(float4*)&Wp[bb][kp * WP_ROW + n4 * 8 + 4] = hi;
        }
    };

    stage(0, 0);
    __syncthreads();

    const int nchunks = D_DIM / KC;   // 24
    for (int j = 0; j < nchunks; ++j) {
        const int cur = j & 1;
        if (j + 1 < nchunks) stage((j + 1) * KC, cur ^ 1);

        #pragma unroll
        for (int kk = 0; kk < KC; kk += 4) {
            // A fragments (16x4): lane r holds row, elems K = kk+2h, kk+2h+1
            v2f va0 = *(const v2f*)&Xs[cur][(wm * 32 + r)      * XS_STRIDE + kk + 2 * h];
            v2f va1 = *(const v2f*)&Xs[cur][(wm * 32 + 16 + r) * XS_STRIDE + kk + 2 * h];
            // B fragments (4x16): one interleaved float2 per lane
            const int kp = (kk >> 1) + h;
            v2f vb0 = *(const v2f*)&Wp[cur][kp * WP_ROW + (wn * 32 + r)      * 2];
            v2f vb1 = *(const v2f*)&Wp[cur][kp * WP_ROW + (wn * 32 + 16 + r) * 2];
            a00 = __builtin_amdgcn_wmma_f32_16x16x4_f32(false, va0, false, vb0,
                                                        (short)0, a00, false, false);
            a01 = __builtin_amdgcn_wmma_f32_16x16x4_f32(false, va0, false, vb1,
                                                        (short)0, a01, false, false);
            a10 = __builtin_amdgcn_wmma_f32_16x16x4_f32(false, va1, false, vb0,
                                                        (short)0, a10, false, false);
            a11 = __builtin_amdgcn_wmma_f32_16x16x4_f32(false, va1, false, vb1,
                                                        (short)0, a11, false, false);
        }
        __syncthreads();
    }

    // Store C: VGPR v, lane l -> M = (l>=16 ? 8 : 0) + v, N = l&15.
    #pragma unroll
    for (int v = 0; v < 8; ++v) {
        const int mlo = m0 + wm * 32 + h * 8 + v;
        const int mhi = mlo + 16;
        const int nc0 = n0 + wn * 32 + r;
        U[(size_t)mlo * NC + nc0]      = a00[v];
        U[(size_t)mlo * NC + nc0 + 16] = a01[v];
        U[(size_t)mhi * NC + nc0]      = a10[v];
        U[(size_t)mhi * NC + nc0 + 16] = a11[v];
    }
}

// ---------------------------------------------------------------------------
// Routing: one 512-thread workgroup per batch element. All 3 iterations.
// u_hat viewed as [S][N][C] per batch (row-major from the GEMM).
// ---------------------------------------------------------------------------
__global__ __launch_bounds__(512)
void caps_routing(const float* __restrict__ U, float* __restrict__ out) {
    __shared__ float cw[NCAP * S_DIM];      // 32 KB: b-logits, then softmax c
    __shared__ float ob[NCAP * CCAP];       //  2 KB: capsule outputs

    const int t = threadIdx.x;              // 0..511
    const int b = blockIdx.x;
    const float* Ub = U + (size_t)b * S_DIM * NC;

    // b starts at zero
    for (int i = t; i < NCAP * S_DIM; i += 512) cw[i] = 0.0f;
    __syncthreads();

    const int n = t >> 5;
    const int c = t & 31;

    for (int iter = 0; iter < 3; ++iter) {
        // --- softmax over the capsule axis (16 values) for column s = t ---
        {
            const int s = t;
            float m = cw[s];
            #pragma unroll
            for (int nn = 1; nn < NCAP; ++nn)
                m = fmaxf(m, cw[nn * S_DIM + s]);
            float sum = 0.0f;
            #pragma unroll
            for (int nn = 0; nn < NCAP; ++nn) {
                float e = __expf(cw[nn * S_DIM + s] - m);
                cw[nn * S_DIM + s] = e;
                sum += e;
            }
            float inv = 1.0f / sum;
            #pragma unroll
            for (int nn = 0; nn < NCAP; ++nn)
                cw[nn * S_DIM + s] *= inv;
        }
        __syncthreads();

        // --- outputs[n][c] = sum_s c[n][s] * u_hat[s][n][c] ---
        float acc = 0.0f;
        {
            const float* up = Ub + n * CCAP + c;     // stride NC over s
            const float* cp = &cw[n * S_DIM];
            for (int s = 0; s < S_DIM; ++s)
                acc += cp[s] * up[(size_t)s * NC];
        }
        ob[t] = acc;
        __syncthreads();

        // --- squash: x / sqrt(sum(x^2) + 1e-7) over the 32-dim capsule ---
        float ssq = 0.0f;
        #pragma unroll
        for (int cc = 0; cc < CCAP; ++cc) {
            float v = ob[n * CCAP + cc];
            ssq += v * v;
        }
        float o = acc / sqrtf(ssq + 1e-7f);
        __syncthreads();
        ob[t] = o;
        __syncthreads();

        if (iter == 2) {
            out[(size_t)b * NC + t] = o;   // [B, N, C]
        } else {
            // --- b[n][s] = dot(outputs[n][:], u_hat[s][n][:]) ---
            for (int p = t; p < NCAP * S_DIM; p += 512) {
                int n2 = p >> 9;
                int s2 = p & (S_DIM - 1);
                const float4* row = (const float4*)(Ub + (size_t)s2 * NC + n2 * CCAP);
                const float4* ov  = (const float4*)&ob[n2 * CCAP];
                float d = 0.0f;
                #pragma unroll
                for (int q = 0; q < 8; ++q) {
                    float4 u = row[q];
                    float4 w = ov[q];
                    d += u.x * w.x + u.y * w.y + u.z * w.z + u.w * w.w;
                }
                cw[n2 * S_DIM + s2] = d;
            }
            __syncthreads();
        }
    }
}

extern "C" void kernel_launch(void* const* d_in, const int* in_sizes, int n_in,
                              void* d_out, int out_size, void* d_ws, size_t ws_size,
                              hipStream_t stream) {
    (void)in_sizes; (void)n_in; (void)out_size; (void)ws_size;
    const float* X = (const float*)d_in[0];   // [32, 512, 768]
    const float* W = (const float*)d_in[1];   // [1, 768, 512]
    float* out  = (float*)d_out;              // [32, 16, 32]
    float* uhat = (float*)d_ws;               // [16384, 512] = 33.5 MB scratch

    dim3 gGemm(NC / NT, M_TOT / MT);          // (8, 128)
    caps_gemm_wmma_f32<<<gGemm, 256, 0, stream>>>(X, W, uhat);
    caps_routing<<<B_DIM, 512, 0, stream>>>(uhat, out);
}